// LabeledConv_86002425135385
// MI455X (gfx1250) — compile-verified
//
#include <hip/hip_runtime.h>

typedef float v2f __attribute__((ext_vector_type(2)));
typedef float v8f __attribute__((ext_vector_type(8)));

#define CH 256   // in == out channels

// ---------------------------------------------------------------------------
// Degree kernels: deg starts at 1.0 (self-loop), accumulate targets, rsqrt.
// deg layout in ws: [4][M]
// ---------------------------------------------------------------------------
__global__ void k_deg_init(float* __restrict__ deg, int M) {
    int i = blockIdx.x * blockDim.x + threadIdx.x;
    if (i < M) {
        deg[i] = 1.0f;
        deg[M + i] = 1.0f;
        deg[2 * (size_t)M + i] = 1.0f;
        deg[3 * (size_t)M + i] = 1.0f;
    }
}

__global__ void k_deg_count(const int* __restrict__ e0, const int* __restrict__ e1,
                            const int* __restrict__ e2, const int* __restrict__ e3,
                            float* __restrict__ deg, int M, int E) {
    int i = blockIdx.x * blockDim.x + threadIdx.x;
    if (i < E) {
        atomicAdd(&deg[e0[E + i]], 1.0f);                      // col = edges[1]
        atomicAdd(&deg[(size_t)M + e1[E + i]], 1.0f);
        atomicAdd(&deg[2 * (size_t)M + e2[E + i]], 1.0f);
        atomicAdd(&deg[3 * (size_t)M + e3[E + i]], 1.0f);
    }
}

__global__ void k_deg_rsqrt(float* __restrict__ deg, int M) {
    int i = blockIdx.x * blockDim.x + threadIdx.x;
    if (i < M) {
        deg[i] = rsqrtf(deg[i]);
        deg[(size_t)M + i] = rsqrtf(deg[(size_t)M + i]);
        deg[2 * (size_t)M + i] = rsqrtf(deg[2 * (size_t)M + i]);
        deg[3 * (size_t)M + i] = rsqrtf(deg[3 * (size_t)M + i]);
    }
}

// ---------------------------------------------------------------------------
// h_i = x @ W_i via V_WMMA_F32_16X16X4_F32 with 2x2 register blocking:
// one wave computes a 32x32 output tile (4 accumulators) so each K-step
// amortizes 2 A-frags + 2 B-frags (1 KB) over 4 WMMAs -> 256 B/WMMA.
// Block = 8 waves = 128x64 of output; grid = (ceil(M/128), CH/64, 4 convs).
// W (256 KB) and x (100 MB) are both L2-resident (192 MB), so the 4x reuse
// of x across blockIdx.y never re-touches HBM.
// ---------------------------------------------------------------------------
__global__ void __launch_bounds__(256)
k_gemm_wmma(const float* __restrict__ x,
            const float* __restrict__ Wa, const float* __restrict__ Wb,
            const float* __restrict__ Wc, const float* __restrict__ Wd,
            float* __restrict__ h, int M) {
    const int cv = blockIdx.z;
    const float* W = (cv == 0) ? Wa : (cv == 1) ? Wb : (cv == 2) ? Wc : Wd;
    float* hout = h + (size_t)cv * M * CH;

    const int wave = threadIdx.x >> 5;            // wave32
    const int lane = threadIdx.x & 31;
    const int wm = wave & 3;                      // 4 wave-rows
    const int wn = wave >> 2;                     // 2 wave-cols
    const int m0 = blockIdx.x * 128 + wm * 32;
    if (m0 >= M) return;                          // wave-uniform; M % 32 == 0
    const int n0 = blockIdx.y * 64 + wn * 32;

    const int l15   = lane & 15;
    const int hi    = lane >> 4;
    const int khalf = hi << 1;                    // lanes 0-15: K+{0,1}; 16-31: K+{2,3}

    // A fragments: rows m0..m0+15 and m0+16..m0+31, float2 per lane
    const float* arow0 = x + (size_t)(m0 + l15) * CH + khalf;
    const float* arow1 = arow0 + 16 * CH;
    // B fragments: cols n0..n0+15 and n0+16..n0+31
    const float* bcol0 = W + (size_t)khalf * CH + (n0 + l15);
    const float* bcol1 = bcol0 + 16;

    v8f c00 = {}, c01 = {}, c10 = {}, c11 = {};
#pragma unroll 4
    for (int kb = 0; kb < CH; kb += 4) {
        v2f a0 = *(const v2f*)(arow0 + kb);
        v2f a1 = *(const v2f*)(arow1 + kb);
        v2f b0, b1;
        b0.x = bcol0[(size_t)kb * CH];
        b0.y = bcol0[(size_t)(kb + 1) * CH];
        b1.x = bcol1[(size_t)kb * CH];
        b1.y = bcol1[(size_t)(kb + 1) * CH];
        // (neg_a, A, neg_b, B, c_mod, C, reuse_a, reuse_b)
        c00 = __builtin_amdgcn_wmma_f32_16x16x4_f32(false, a0, false, b0, (short)0, c00, false, false);
        c01 = __builtin_amdgcn_wmma_f32_16x16x4_f32(false, a0, false, b1, (short)0, c01, false, false);
        c10 = __builtin_amdgcn_wmma_f32_16x16x4_f32(false, a1, false, b0, (short)0, c10, false, false);
        c11 = __builtin_amdgcn_wmma_f32_16x16x4_f32(false, a1, false, b1, (short)0, c11, false, false);
    }

    // C/D layout: VGPR v -> row base+v (lanes 0-15) / base+8+v (lanes 16-31)
    const int rbase = m0 + (hi << 3);
    float* hp0 = hout + (size_t)rbase * CH + (n0 + l15);         // C00
    float* hp1 = hp0 + 16;                                        // C01
    float* hp2 = hp0 + 16 * CH;                                   // C10
    float* hp3 = hp2 + 16;                                        // C11
#pragma unroll
    for (int v = 0; v < 8; ++v) {
        hp0[(size_t)v * CH] = c00[v];
        hp1[(size_t)v * CH] = c01[v];
        hp2[(size_t)v * CH] = c10[v];
        hp3[(size_t)v * CH] = c11[v];
    }
}

// ---------------------------------------------------------------------------
// out[n][c] = t0*x[n,0]*p0 + t1*x[n,0]*p1
//           + sum_i p_{i+2}[c] * ( b_i[c] + h_i[n][c] / deg_i[n] )   (self-loop)
// One block per node, one thread per channel.
// ---------------------------------------------------------------------------
__global__ void __launch_bounds__(CH)
k_init_out(const float* __restrict__ x,
           const float* __restrict__ t0, const float* __restrict__ t1,
           const float* __restrict__ p0, const float* __restrict__ p1,
           const float* __restrict__ p2, const float* __restrict__ p3,
           const float* __restrict__ p4, const float* __restrict__ p5,
           const float* __restrict__ b1, const float* __restrict__ b2,
           const float* __restrict__ b3, const float* __restrict__ b4,
           const float* __restrict__ h, const float* __restrict__ dis,
           float* __restrict__ out, int M) {
    const int n = blockIdx.x;
    const int c = threadIdx.x;
    const float xc = x[(size_t)n * CH];

    float acc = t0[n] * xc * p0[c] + t1[n] * xc * p1[c];

    float d;
    d = dis[n];                     // dis = rsqrt(deg) -> d*d = 1/deg
    acc += p2[c] * (b1[c] + h[(size_t)n * CH + c] * d * d);
    d = dis[(size_t)M + n];
    acc += p3[c] * (b2[c] + h[((size_t)M * CH) + (size_t)n * CH + c] * d * d);
    d = dis[2 * (size_t)M + n];
    acc += p4[c] * (b3[c] + h[(2 * (size_t)M * CH) + (size_t)n * CH + c] * d * d);
    d = dis[3 * (size_t)M + n];
    acc += p5[c] * (b4[c] + h[(3 * (size_t)M * CH) + (size_t)n * CH + c] * d * d);

    out[(size_t)n * CH + c] = acc;
}

// ---------------------------------------------------------------------------
// Edge scatter with the per-channel p-scale folded in:
//   out[col][c] += p_{cv+2}[c] * dis[row]*dis[col] * h_cv[row][c]
// One wave per edge: 32 lanes x 8 channels, float4 loads, f32 atomics.
// grid = (ceil(E/8), 4), block = 256.
// ---------------------------------------------------------------------------
__global__ void __launch_bounds__(256)
k_scatter(const int* __restrict__ e0, const int* __restrict__ e1,
          const int* __restrict__ e2, const int* __restrict__ e3,
          const float* __restrict__ p2, const float* __restrict__ p3,
          const float* __restrict__ p4, const float* __restrict__ p5,
          const float* __restrict__ h, const float* __restrict__ dis,
          float* __restrict__ out, int M, int E) {
    const int cv = blockIdx.y;
    const int* er = (cv == 0) ? e0 : (cv == 1) ? e1 : (cv == 2) ? e2 : e3;
    const float* ps = (cv == 0) ? p2 : (cv == 1) ? p3 : (cv == 2) ? p4 : p5;
    const float* hh = h + (size_t)cv * M * CH;
    const float* ds = dis + (size_t)cv * M;

    const int eg = blockIdx.x * 8 + (threadIdx.x >> 5);
    if (eg >= E) return;
    const int lane = threadIdx.x & 31;

    const int row = er[eg];          // source
    const int col = er[E + eg];      // target
    const float norm = ds[row] * ds[col];

    const float* hp = hh + (size_t)row * CH + lane * 8;
    const float* pp = ps + lane * 8;
    float*       op = out + (size_t)col * CH + lane * 8;

    const float4 ha = *(const float4*)(hp);
    const float4 hb = *(const float4*)(hp + 4);
    const float4 pa = *(const float4*)(pp);
    const float4 pb = *(const float4*)(pp + 4);

    atomicAdd(op + 0, norm * ha.x * pa.x);
    atomicAdd(op + 1, norm * ha.y * pa.y);
    atomicAdd(op + 2, norm * ha.z * pa.z);
    atomicAdd(op + 3, norm * ha.w * pa.w);
    atomicAdd(op + 4, norm * hb.x * pb.x);
    atomicAdd(op + 5, norm * hb.y * pb.y);
    atomicAdd(op + 6, norm * hb.z * pb.z);
    atomicAdd(op + 7, norm * hb.w * pb.w);
}

// ---------------------------------------------------------------------------
extern "C" void kernel_launch(void* const* d_in, const int* in_sizes, int n_in,
                              void* d_out, int out_size, void* d_ws, size_t ws_size,
                              hipStream_t stream) {
    const float* x   = (const float*)d_in[0];
    const int*   e00 = (const int*)d_in[1];
    const int*   e01 = (const int*)d_in[2];
    const int*   e10 = (const int*)d_in[3];
    const int*   e11 = (const int*)d_in[4];
    const float* t0  = (const float*)d_in[5];
    const float* t1  = (const float*)d_in[6];
    const float* W1  = (const float*)d_in[7];
    const float* b1  = (const float*)d_in[8];
    const float* W2  = (const float*)d_in[9];
    const float* b2  = (const float*)d_in[10];
    const float* W3  = (const float*)d_in[11];
    const float* b3  = (const float*)d_in[12];
    const float* W4  = (const float*)d_in[13];
    const float* b4  = (const float*)d_in[14];
    const float* p0  = (const float*)d_in[15];
    const float* p1  = (const float*)d_in[16];
    const float* p2  = (const float*)d_in[17];
    const float* p3  = (const float*)d_in[18];
    const float* p4  = (const float*)d_in[19];
    const float* p5  = (const float*)d_in[20];

    const int M = in_sizes[0] / CH;   // 100000 nodes
    const int E = in_sizes[1] / 2;    // 500000 edges per set

    // Workspace layout: [4*M] deg/dis floats, then [4*M*CH] h floats (~411 MB).
    float* deg = (float*)d_ws;
    float* h   = deg + 4 * (size_t)M;
    float* out = (float*)d_out;

    // 1) degrees -> rsqrt
    k_deg_init<<<(M + 255) / 256, 256, 0, stream>>>(deg, M);
    k_deg_count<<<(E + 255) / 256, 256, 0, stream>>>(e00, e01, e10, e11, deg, M, E);
    k_deg_rsqrt<<<(M + 255) / 256, 256, 0, stream>>>(deg, M);

    // 2) h_i = x @ W_i  (WMMA f32 16x16x4, 2x2 register-blocked)
    dim3 gg((M + 127) / 128, CH / 64, 4);
    k_gemm_wmma<<<gg, 256, 0, stream>>>(x, W1, W2, W3, W4, h, M);

    // 3) init out with rank-1 terms + biases + self-loop contributions
    k_init_out<<<M, CH, 0, stream>>>(x, t0, t1, p0, p1, p2, p3, p4, p5,
                                     b1, b2, b3, b4, h, deg, out, M);

    // 4) edge scatter with folded per-channel scales
    dim3 gs((E + 7) / 8, 4);
    k_scatter<<<gs, 4 * 64, 0, stream>>>(e00, e01, e10, e11, p2, p3, p4, p5,
                                         h, deg, out, M, E);
}